// TakeMedian_5463198401147
// MI455X (gfx1250) — compile-verified
//
#include <hip/hip_runtime.h>

#ifndef __has_builtin
#define __has_builtin(x) 0
#endif

// Problem constants (match reference: B=32, T=4096, D=512)
#define B_      32
#define T_      4096
#define D_      512

// Tiling: one block handles (batch b, 32 consecutive features).
#define COLS    32                 // features per block
#define TPC     8                  // threads cooperating per feature column
#define NTHREADS (COLS * TPC)      // 256 threads = 8 wave32
#define TILE_T  64                 // time rows staged per LDS tile buffer
#define NBINS   256                // 8-bit radix digit
#define CHUNKS  2                  // 16B async chunks issued per thread per tile

#define HIST_WORDS (NBINS * COLS)          // 8192 u32 = 32 KB
#define TILE_WORDS (TILE_T * COLS)         // 2048 f32 =  8 KB (x2 buffers)
#define SMEM_BYTES ((HIST_WORDS + 2 * TILE_WORDS + 2 * COLS) * 4)

#if __has_builtin(__builtin_amdgcn_global_load_async_to_lds_b128)
#define HAVE_ASYNC 1
// Builtin signature (from hipcc diagnostic): arg0 = int4 addrspace(1)*, arg1 = int4 addrspace(3)*
typedef int v4i_t __attribute__((ext_vector_type(4)));
typedef __attribute__((address_space(1))) v4i_t* gv4_ptr;
typedef __attribute__((address_space(3))) v4i_t* lv4_ptr;
#endif

#if __has_builtin(__builtin_amdgcn_s_wait_asynccnt)
#define WAIT_ASYNC(n) __builtin_amdgcn_s_wait_asynccnt(n)
#else
#define WAIT_ASYNC(n) asm volatile("s_wait_asynccnt %0" ::"i"(n) : "memory")
#endif

__global__ __launch_bounds__(NTHREADS)
void TakeMedian_radixsel_kernel(const float* __restrict__ in,
                                const int* __restrict__ lens,
                                float* __restrict__ out) {
    extern __shared__ unsigned char smem[];
    unsigned int* hist   = (unsigned int*)smem;                              // [NBINS][COLS]
    float*        tile   = (float*)(smem + HIST_WORDS * 4);                  // [2][TILE_T][COLS]
    unsigned int* s_pref = (unsigned int*)(smem + (HIST_WORDS + 2 * TILE_WORDS) * 4);
    unsigned int* s_kth  = s_pref + COLS;

    const int tid   = threadIdx.x;
    const int col   = tid % COLS;       // feature column owned (shared by TPC threads)
    const int tsub  = tid / COLS;       // 0..TPC-1 : which wave of the column team
    const int b     = blockIdx.y;
    const int dbase = blockIdx.x * COLS;
    const int len   = lens[b];          // uniform across the block (per-batch ragged length)

    if (tid < COLS) {
        s_pref[tid] = 0u;
        s_kth[tid]  = (unsigned int)((len - 1) >> 1);   // lower-median rank
    }

    // Staging assignment: CHUNKS consecutive 16B chunks per thread.
    // Tile row = COLS*4 = 128 B = 8 chunks; TILE_T*8 = 512 chunks / 256 threads = 2 each.
    const int    q0   = tid * CHUNKS;
    const int    srow = q0 >> 3;            // row in tile
    const int    scol = (q0 & 7) * 4;       // float offset in row (16B aligned)
    const size_t gBase = ((size_t)b * T_) * D_ + (size_t)dbase + (size_t)scol;

    const int nt = (len + TILE_T - 1) / TILE_T;   // number of tiles (uniform per block)

    // Issue async DMA for one tile into buffer `buf`.
    auto stage = [&](int buf, int t0) {
        const float* src = in + gBase + (size_t)(t0 + srow) * D_;
        float*       dst = tile + buf * TILE_WORDS + srow * COLS + scol;
#if HAVE_ASYNC
#pragma unroll
        for (int c = 0; c < CHUNKS; ++c) {
            __builtin_amdgcn_global_load_async_to_lds_b128(
                (gv4_ptr)(unsigned long long)(src + 4 * c), (lv4_ptr)(dst + 4 * c), 0, 0);
        }
#else
#pragma unroll
        for (int c = 0; c < CHUNKS; ++c) {
            *(float4*)(dst + 4 * c) = *(const float4*)(src + 4 * c);
        }
#endif
    };

    __syncthreads();

    // 4-pass radix select over the sign-flipped float bit pattern (MSB first).
    for (int pass = 0; pass < 4; ++pass) {
        const int shift = 24 - 8 * pass;
        for (int i = tid; i < HIST_WORDS; i += NTHREADS) hist[i] = 0u;

        stage(0, 0);              // prologue: DMA first tile while zeroing finishes
        __syncthreads();

        const unsigned int prefHi = s_pref[col];
        const unsigned int maskHi = pass ? (0xFFFFFFFFu << (shift + 8)) : 0u;

        for (int it = 0; it < nt; ++it) {
            const int cur = it & 1;
            if (it + 1 < nt) {
                stage(cur ^ 1, (it + 1) * TILE_T);  // prefetch next tile (other buffer)
                WAIT_ASYNC(CHUNKS);                 // oldest CHUNKS (current tile) landed
            } else {
                WAIT_ASYNC(0);                      // last tile: drain everything
            }
            __syncthreads();                        // all waves' DMA for `cur` visible

            // ---- rank elements of this tile into the per-column histogram ----
            const int t0   = it * TILE_T;
            const int rows = min(TILE_T, len - t0);
            const float* tbuf = tile + cur * TILE_WORDS;
            for (int r = tsub; r < rows; r += TPC) {
                unsigned int u = __float_as_uint(tbuf[r * COLS + col]);
                u = (u & 0x80000000u) ? ~u : (u | 0x80000000u);   // order-preserving flip
                if ((u & maskHi) == (prefHi & maskHi)) {
                    atomicAdd(&hist[(((u >> shift) & 255u) * COLS) + col], 1u);
                }
            }
            __syncthreads();   // everyone done reading `cur` before it is restaged
        }

        // ---- per-column digit scan: find bin containing rank k ----
        if (tid < COLS) {
            unsigned int k = s_kth[col];
            unsigned int cum = 0;
            int bin = NBINS - 1;
            for (int i = 0; i < NBINS; ++i) {
                unsigned int c = hist[i * COLS + col];
                if (cum + c > k) { bin = i; break; }
                cum += c;
            }
            s_kth[col]  = k - cum;
            s_pref[col] = prefHi | ((unsigned int)bin << shift);
        }
        __syncthreads();
    }

    // ---- write result: undo the order flip, append len as float ----
    if (tid < COLS) {
        const unsigned int u = s_pref[col];
        const unsigned int bits = (u & 0x80000000u) ? (u ^ 0x80000000u) : ~u;
        out[(size_t)b * (D_ + 1) + dbase + col] = __uint_as_float(bits);
    }
    if (tid == 0 && blockIdx.x == 0) {
        out[(size_t)b * (D_ + 1) + D_] = (float)len;
    }
}

extern "C" void kernel_launch(void* const* d_in, const int* in_sizes, int n_in,
                              void* d_out, int out_size, void* d_ws, size_t ws_size,
                              hipStream_t stream) {
    (void)in_sizes; (void)n_in; (void)out_size; (void)d_ws; (void)ws_size;
    const float* in   = (const float*)d_in[0];
    const int*   lens = (const int*)d_in[1];
    float*       out  = (float*)d_out;

    dim3 grid(D_ / COLS, B_);      // (16, 32) blocks
    dim3 block(NTHREADS);          // 256 threads = 8 wave32
    TakeMedian_radixsel_kernel<<<grid, block, SMEM_BYTES, stream>>>(in, lens, out);
}